// OGQC_65386582114672
// MI455X (gfx1250) — compile-verified
//
#include <hip/hip_runtime.h>
#include <math.h>

#define NUM_BINS 36
#define NB2 (NUM_BINS * NUM_BINS)      // 1296
#define HID 256
#define BB 16
#define QQ 900
#define NT 57                           // ceil(900/16)
#define SORTN 2048
#define BIGV 1e9f
#define TWO_PI_F 6.2831853071795864f
#define PI_F 3.14159265358979323f
#define BIN_SIZE_F (TWO_PI_F / NUM_BINS)

typedef float v2f __attribute__((ext_vector_type(2)));
typedef float v8f __attribute__((ext_vector_type(8)));

// ---------- helpers ----------
__device__ __forceinline__ unsigned int ford(float f) {
    unsigned int u = __float_as_uint(f);
    return (u & 0x80000000u) ? ~u : (u | 0x80000000u);
}
__device__ __forceinline__ float funord(unsigned int o) {
    unsigned int u = (o & 0x80000000u) ? (o ^ 0x80000000u) : ~o;
    return __uint_as_float(u);
}
__device__ __forceinline__ float circ_dist(float a, float b) {
    float diff = fabsf(a - b);
    return fminf(fminf(diff, TWO_PI_F - diff), PI_F * 0.5f);
}
__device__ __forceinline__ void top4_insert(float v, float& a0, float& a1, float& a2, float& a3) {
    if (v > a3) {
        if (v > a0)      { a3 = a2; a2 = a1; a1 = a0; a0 = v; }
        else if (v > a1) { a3 = a2; a2 = a1; a1 = v; }
        else if (v > a2) { a3 = a2; a2 = v; }
        else               a3 = v;
    }
}

// ---------- kernel 0: init atomics / strip-top scratch ----------
__global__ __launch_bounds__(256) void init_kernel(unsigned long long* __restrict__ segpacked,
                                                   float* __restrict__ striptop) {
    int idx = blockIdx.x * 256 + threadIdx.x;
    if (idx < BB * NB2) segpacked[idx] = 0ull;
    if (idx < BB * NT * 4) striptop[idx] = -BIGV;
}

// ---------- kernel 1: angles, bins, fused + L2-normalize ----------
__global__ __launch_bounds__(256) void prep_kernel(const float* __restrict__ qf,
                                                   const float* __restrict__ pa,
                                                   const float* __restrict__ W,
                                                   float* __restrict__ nq,
                                                   float* __restrict__ angles,
                                                   int* __restrict__ binsA) {
    const int bq = blockIdx.x;                  // b*QQ + q
    const int t  = threadIdx.x;
    float y = pa[(size_t)bq * 2 + 1];
    float x = pa[(size_t)bq * 2 + 0];
    float ang = atan2f(y, x);
    if (ang < 0.f) ang += TWO_PI_F;             // python: % 2pi
    int bin = (int)(ang / BIN_SIZE_F);
    bin = bin < 0 ? 0 : (bin > NUM_BINS - 1 ? NUM_BINS - 1 : bin);

    float f = qf[(size_t)bq * HID + t] + W[(size_t)bin * HID + t];
    __shared__ float red[256];
    red[t] = f * f;
    __syncthreads();
    for (int off = 128; off > 0; off >>= 1) {
        if (t < off) red[t] += red[t + off];
        __syncthreads();
    }
    float nrm = fmaxf(sqrtf(red[0]), 1e-12f);
    nq[(size_t)bq * HID + t] = f / nrm;
    if (t == 0) { angles[bq] = ang; binsA[bq] = bin; }
}

// ---------- kernel 2: WMMA fp32 row-strip GEMM ----------
// One workgroup (8 waves) per (batch, 16-row strip ti). Each wave computes the
// 16x16 tiles tj = ti + wave, ti + wave + 8, ... with v_wmma_f32_16x16x4_f32.
// Per-bin-pair argmax is aggregated in LDS (ds_max_u64) and flushed once per
// workgroup; per-strip top-4 feeds the exact global top-8 threshold.
__global__ __launch_bounds__(256) void sim_strip_kernel(const float* __restrict__ nq,
                                                        const int* __restrict__ binsA,
                                                        unsigned long long* __restrict__ segpacked,
                                                        float* __restrict__ striptop) {
    const int ti = blockIdx.x;                  // 0..NT-1 (row strip)
    const int b  = blockIdx.y;
    const int t    = threadIdx.x;
    const int wave = t >> 5;                    // 0..7
    const int lane = t & 31;                    // wave32
    const int half = lane >> 4;                 // 0 or 1
    const int l15  = lane & 15;
    const int i0   = ti * 16;

    __shared__ unsigned long long lseg[NB2];    // per-WG segment argmax table
    __shared__ float tbuf[1024];                // top-4 reduce scratch

    for (int s = t; s < NB2; s += 256) lseg[s] = 0ull;
    __syncthreads();

    const float* base = nq + (size_t)b * QQ * HID;
    const float* pA = base + (size_t)min(i0 + l15, QQ - 1) * HID;   // A rows (M = l15)
    const int ko = 2 * half;                    // per-lane K-pair (ISA 32-bit A/B layout)

    float lt0 = -BIGV, lt1 = -BIGV, lt2 = -BIGV, lt3 = -BIGV;

    for (int tj = ti + wave; tj < NT; tj += 8) {        // uniform per wave
        const int j0 = tj * 16;
        const float* pB = base + (size_t)min(j0 + l15, QQ - 1) * HID;  // B cols (N = l15)

        v8f c = {0.f, 0.f, 0.f, 0.f, 0.f, 0.f, 0.f, 0.f};
        for (int kk = 0; kk < HID; kk += 4) {
            v2f a, bf;
            a.x  = pA[kk + ko];  a.y  = pA[kk + ko + 1];
            bf.x = pB[kk + ko];  bf.y = pB[kk + ko + 1];
            // D = A(16x4,f32) * B(4x16,f32) + C  -> v_wmma_f32_16x16x4_f32
            c = __builtin_amdgcn_wmma_f32_16x16x4_f32(false, a, false, bf, (short)0, c,
                                                      false, false);
        }

        const int j  = j0 + l15;
        const int bj = binsA[b * QQ + min(j, QQ - 1)];
#pragma unroll
        for (int r = 0; r < 8; r++) {
            const int i = i0 + r + 8 * half;    // C/D layout: VGPR r -> M = r + 8*half
            const float v = c[r];
            if (i < j && j < QQ) {
                const int bi = binsA[b * QQ + i];
                const int s  = (bi < bj ? bi : bj) * NUM_BINS + (bi < bj ? bj : bi);
                // max sim; ties -> min (i,j) lex == first occurrence in triu order
                unsigned long long pk = ((unsigned long long)ford(v) << 32)
                                      | (unsigned long long)(0xFFFFFFFFu - (unsigned)(i * 1024 + j));
                atomicMax(&lseg[s], pk);        // LDS ds_max_u64 (low contention)
                top4_insert(v, lt0, lt1, lt2, lt3);
            }
        }
    }

    __syncthreads();
    // flush per-WG segment maxima: <=1296 global atomics per workgroup
    for (int s = t; s < NB2; s += 256) {
        unsigned long long p = lseg[s];
        if (p != 0ull) atomicMax(&segpacked[(size_t)b * NB2 + s], p);
    }

    // per-strip top-4
    tbuf[t * 4 + 0] = lt0; tbuf[t * 4 + 1] = lt1;
    tbuf[t * 4 + 2] = lt2; tbuf[t * 4 + 3] = lt3;
    __syncthreads();
    if (t == 0) {
        float g0 = -BIGV, g1 = -BIGV, g2 = -BIGV, g3 = -BIGV;
        for (int k = 0; k < 1024; k++) top4_insert(tbuf[k], g0, g1, g2, g3);
        float* dst = striptop + ((size_t)b * NT + ti) * 4;
        dst[0] = g0; dst[1] = g1; dst[2] = g2; dst[3] = g3;
    }
}

// ---------- tail helpers ----------
__device__ void bitonic_sort_2048(float* buf) {   // ascending; 256 threads
    for (int k = 2; k <= SORTN; k <<= 1) {
        for (int j = k >> 1; j > 0; j >>= 1) {
            for (int idx = threadIdx.x; idx < SORTN; idx += 256) {
                int ixj = idx ^ j;
                if (ixj > idx) {
                    float a = buf[idx], bb = buf[ixj];
                    bool up = ((idx & k) == 0);
                    if (up ? (a > bb) : (a < bb)) { buf[idx] = bb; buf[ixj] = a; }
                }
            }
            __syncthreads();
        }
    }
}
__device__ __forceinline__ float quantile_sorted(const float* sbuf, int n) {
    float nf = (float)n;
    float q  = 1.0f - 1.0f / (nf + 1.0f);
    float pos = q * fmaxf(nf - 1.0f, 0.0f);
    int lo = (int)floorf(pos); lo = lo < 0 ? 0 : (lo > NB2 - 1 ? NB2 - 1 : lo);
    int hi = (int)ceilf(pos);  hi = hi < 0 ? 0 : (hi > NB2 - 1 ? NB2 - 1 : hi);
    float frac = pos - (float)lo;
    return sbuf[lo] + (sbuf[hi] - sbuf[lo]) * frac;
}

// ---------- kernel 3: per-batch quantiles / median / losses ----------
__global__ __launch_bounds__(256) void tail_kernel(const unsigned long long* __restrict__ segpacked,
                                                   const float* __restrict__ angles,
                                                   const float* __restrict__ striptop,
                                                   float* __restrict__ coopB,
                                                   float* __restrict__ compB,
                                                   float* __restrict__ vbB) {
    const int b = blockIdx.x;
    const int t = threadIdx.x;

    __shared__ float ssim[NB2];
    __shared__ float dd[NB2];
    __shared__ float aiA[NB2], ajA[NB2];
    __shared__ unsigned char validA[NB2];
    __shared__ unsigned char maskFA[NB2];
    __shared__ float sortbuf[SORTN];
    __shared__ int s_cnt, s_cnt2;
    __shared__ float s_sum, s_sum2;

    // unpack segments
    for (int s = t; s < NB2; s += 256) {
        unsigned long long p = segpacked[(size_t)b * NB2 + s];
        bool valid = (p != 0ull);
        validA[s] = valid ? 1 : 0;
        if (valid) {
            float v = funord((unsigned int)(p >> 32));
            unsigned int key = 0xFFFFFFFFu - (unsigned int)(p & 0xFFFFFFFFu);
            int i = key >> 10, j = key & 1023;
            float ai = angles[b * QQ + i], aj = angles[b * QQ + j];
            ssim[s] = v; aiA[s] = ai; ajA[s] = aj; dd[s] = circ_dist(ai, aj);
        } else { ssim[s] = BIGV; dd[s] = BIGV; aiA[s] = 0.f; ajA[s] = 0.f; }
    }
    __syncthreads();

    // S = #valid
    if (t == 0) s_cnt = 0;
    __syncthreads();
    { int lc = 0; for (int s = t; s < NB2; s += 256) lc += validA[s]; atomicAdd(&s_cnt, lc); }
    __syncthreads();
    const int Sv = s_cnt;

    // thr = masked quantile over valid sel_sim
    for (int idx = t; idx < SORTN; idx += 256) sortbuf[idx] = (idx < NB2) ? ssim[idx] : BIGV;
    __syncthreads();
    bitonic_sort_2048(sortbuf);
    const float thr = quantile_sorted(sortbuf, Sv);

    // mask1
    if (t == 0) s_cnt = 0;
    __syncthreads();
    { int lc = 0;
      for (int s = t; s < NB2; s += 256) {
          bool m = validA[s] && (ssim[s] > thr);
          maskFA[s] = m ? 1 : 0; lc += m;
      }
      atomicAdd(&s_cnt, lc);
    }
    __syncthreads();
    const int m1 = s_cnt;

    // th2 = 4th-largest pair sim (== 8th-largest of full symmetric matrix)
    {
        float l0 = -BIGV, l1 = -BIGV, l2 = -BIGV, l3 = -BIGV;
        const float* tb = striptop + (size_t)b * NT * 4;
        for (int idx = t; idx < NT * 4; idx += 256) top4_insert(tb[idx], l0, l1, l2, l3);
        __syncthreads();   // sortbuf reads from thr phase are done
        sortbuf[t * 4 + 0] = l0; sortbuf[t * 4 + 1] = l1;
        sortbuf[t * 4 + 2] = l2; sortbuf[t * 4 + 3] = l3;
        sortbuf[1024 + t * 4 + 0] = -BIGV; sortbuf[1024 + t * 4 + 1] = -BIGV;
        sortbuf[1024 + t * 4 + 2] = -BIGV; sortbuf[1024 + t * 4 + 3] = -BIGV;
        __syncthreads();
        bitonic_sort_2048(sortbuf);
    }
    const float th2 = sortbuf[SORTN - 4];

    // maskF (+ cnt)
    const bool use_fb = (m1 < 2);
    __syncthreads();
    if (t == 0) s_cnt = 0;
    __syncthreads();
    { int lc = 0;
      for (int s = t; s < NB2; s += 256) {
          bool mf = use_fb ? (validA[s] && (ssim[s] >= th2)) : (maskFA[s] != 0);
          maskFA[s] = mf ? 1 : 0; lc += mf;
      }
      atomicAdd(&s_cnt, lc);
    }
    __syncthreads();
    const int cnt = s_cnt;

    // median of d over maskF  (torch lower-middle: index (cnt-1)//2, python -1//2 clipped to 0)
    for (int idx = t; idx < SORTN; idx += 256)
        sortbuf[idx] = (idx < NB2 && maskFA[idx]) ? dd[idx] : BIGV;
    __syncthreads();
    bitonic_sort_2048(sortbuf);
    int mi = (cnt > 0) ? (cnt - 1) / 2 : 0;
    mi = mi > NB2 - 1 ? NB2 - 1 : mi;
    const float med = sortbuf[mi];
    const float alpha = fminf(fmaxf(med, PI_F / NUM_BINS), PI_F * 0.5f);

    // cooperative loss + comp count
    __syncthreads();
    if (t == 0) { s_cnt = 0; s_cnt2 = 0; s_sum = 0.f; }
    __syncthreads();
    { int nco = 0, ncm = 0; float sco = 0.f;
      for (int s = t; s < NB2; s += 256) {
          if (maskFA[s]) {
              float d = dd[s];
              if (d <= alpha) {
                  nco++;
                  float ai = aiA[s], aj = ajA[s];
                  float ma = atan2f((sinf(ai) + sinf(aj)) * 0.5f,
                                    (cosf(ai) + cosf(aj)) * 0.5f);
                  float c1 = circ_dist(ai, ma), c2 = circ_dist(aj, ma);
                  sco += c1 * c1 + c2 * c2;
              } else ncm++;
          }
      }
      atomicAdd(&s_cnt, nco); atomicAdd(&s_cnt2, ncm); atomicAdd(&s_sum, sco);
    }
    __syncthreads();
    const int n_coop = s_cnt, n_comp = s_cnt2;
    const float coop_sum = s_sum;

    // competitive margin quantile over comp_m sims
    for (int idx = t; idx < SORTN; idx += 256)
        sortbuf[idx] = (idx < NB2 && maskFA[idx] && dd[idx] > alpha) ? ssim[idx] : BIGV;
    __syncthreads();
    bitonic_sort_2048(sortbuf);
    const float margin = quantile_sorted(sortbuf, n_comp);

    __syncthreads();
    if (t == 0) s_sum2 = 0.f;
    __syncthreads();
    { float sc2 = 0.f;
      for (int s = t; s < NB2; s += 256) {
          if (maskFA[s] && dd[s] > alpha) {
              float v = fmaxf(ssim[s] - margin, 0.f);
              sc2 += v * v;
          }
      }
      atomicAdd(&s_sum2, sc2);
    }
    __syncthreads();

    if (t == 0) {
        coopB[b] = coop_sum / (float)(n_coop > 1 ? n_coop : 1);
        compB[b] = s_sum2   / (float)(n_comp > 1 ? n_comp : 1);
        vbB[b]   = (cnt > 0) ? 1.f : 0.f;
    }
}

// ---------- kernel 4: combine batches ----------
__global__ void final_kernel(const float* __restrict__ coopB, const float* __restrict__ compB,
                             const float* __restrict__ vbB, float* __restrict__ out) {
    if (threadIdx.x == 0 && blockIdx.x == 0) {
        float sc = 0.f, sm = 0.f; int nv = 0;
        for (int b = 0; b < BB; b++) { sc += coopB[b]; sm += compB[b]; nv += (vbB[b] > 0.5f); }
        float den = (float)(nv > 1 ? nv : 1);
        out[0] = sc / den;
        out[1] = sm / den;
    }
}

// ---------- launch ----------
extern "C" void kernel_launch(void* const* d_in, const int* in_sizes, int n_in,
                              void* d_out, int out_size, void* d_ws, size_t ws_size,
                              hipStream_t stream) {
    (void)in_sizes; (void)n_in; (void)out_size; (void)ws_size;
    const float* qf = (const float*)d_in[0];   // [B,Q,H]
    const float* pa = (const float*)d_in[1];   // [B,Q,2]
    const float* W  = (const float*)d_in[2];   // [36,H]
    float* out = (float*)d_out;                // 2 scalars

    char* ws = (char*)d_ws;
    constexpr size_t SZ_NQ   = (size_t)BB * QQ * HID * 4;
    constexpr size_t SZ_ANG  = (size_t)BB * QQ * 4;
    constexpr size_t SZ_BINS = (size_t)BB * QQ * 4;
    constexpr size_t SZ_SEG  = (size_t)BB * NB2 * 8;
    constexpr size_t SZ_TT   = (size_t)BB * NT * 4 * 4;
    constexpr size_t OFF_NQ   = 0;
    constexpr size_t OFF_ANG  = OFF_NQ + SZ_NQ;
    constexpr size_t OFF_BINS = OFF_ANG + SZ_ANG;
    constexpr size_t OFF_SEG  = OFF_BINS + SZ_BINS;     // 8-byte aligned
    constexpr size_t OFF_TT   = OFF_SEG + SZ_SEG;
    constexpr size_t OFF_CO   = OFF_TT + SZ_TT;
    constexpr size_t OFF_CM   = OFF_CO + BB * 4;
    constexpr size_t OFF_VB   = OFF_CM + BB * 4;

    float* nq      = (float*)(ws + OFF_NQ);
    float* angles  = (float*)(ws + OFF_ANG);
    int*   binsA   = (int*)(ws + OFF_BINS);
    unsigned long long* segpacked = (unsigned long long*)(ws + OFF_SEG);
    float* striptop = (float*)(ws + OFF_TT);
    float* coopB   = (float*)(ws + OFF_CO);
    float* compB   = (float*)(ws + OFF_CM);
    float* vbB     = (float*)(ws + OFF_VB);

    const int init_elems = (BB * NB2 > BB * NT * 4) ? BB * NB2 : BB * NT * 4;
    init_kernel<<<(init_elems + 255) / 256, 256, 0, stream>>>(segpacked, striptop);
    prep_kernel<<<BB * QQ, 256, 0, stream>>>(qf, pa, W, nq, angles, binsA);
    sim_strip_kernel<<<dim3(NT, BB), 256, 0, stream>>>(nq, binsA, segpacked, striptop);
    tail_kernel<<<BB, 256, 0, stream>>>(segpacked, angles, striptop, coopB, compB, vbB);
    final_kernel<<<1, 32, 0, stream>>>(coopB, compB, vbB, out);
}